// AMM_w_AFDM_26508538151422
// MI455X (gfx1250) — compile-verified
//
#include <hip/hip_runtime.h>
#include <math.h>

#define B_   8
#define N_   4096
#define C_   64
#define H_   128
#define W_   128
#define HW_  (H_*W_)
#define MD_  5
#define K_   121
#define EPS_ 1e-5f

typedef __attribute__((ext_vector_type(2))) float v2f;
typedef __attribute__((ext_vector_type(8))) float v8f;

// ---- workspace layout (float offsets) ----
#define OFF_SF     0          // [B*HW] raw channel-sum
#define OFF_MNMX   131072     // 16 uints: [b*2]=min_enc, [b*2+1]=max_enc
#define OFF_STATS  131088     // 64: [b*8 + g*2 + {0:sum,1:sumsq}]
#define OFF_RRAW   131152     // [B*16*HW] conv+bias output
#define OFF_POOL   2228304    // [B*4096] pooled
#define OFF_MNF    2261072    // [b*2]=mn, [b*2+1]=1/(mx-mn+eps)
#define OFF_GNF    2261088    // [b*8+g*2]=mu, +1=rsigma
#define OFF_DIL    2261152    // [B]
#define OFF_WKER   2261160    // [B*121]

// order-preserving float -> uint mapping (for atomic min/max on any-sign floats)
__device__ __forceinline__ unsigned encf(float f) {
  unsigned u = __float_as_uint(f);
  return (u & 0x80000000u) ? ~u : (u | 0x80000000u);
}
__device__ __forceinline__ float decf(unsigned e) {
  return (e & 0x80000000u) ? __uint_as_float(e ^ 0x80000000u)
                           : __uint_as_float(~e);
}

__global__ void k_init(float* ws) {
  int t = threadIdx.x;
  unsigned* mm = (unsigned*)(ws + OFF_MNMX);
  if (t < B_) { mm[t*2] = 0xFFFFFFFFu; mm[t*2+1] = 0u; }
  if (t < 64) ws[OFF_STATS + t] = 0.f;
}

// sf = sum_c feature ; per-batch min/max via block reduce + global atomics
__global__ void k_sf(const float* __restrict__ feat, float* ws) {
  int gid = blockIdx.x * 256 + threadIdx.x;      // B*HW threads; block stays in one b
  int b = gid / HW_, hw = gid % HW_;
  const float* fb = feat + (size_t)b * C_ * HW_ + hw;
  float s = 0.f;
  #pragma unroll 8
  for (int c = 0; c < C_; ++c) s += fb[(size_t)c * HW_];
  ws[OFF_SF + gid] = s;

  __shared__ unsigned smin[256], smax[256];
  unsigned e = encf(s);
  smin[threadIdx.x] = e; smax[threadIdx.x] = e;
  __syncthreads();
  for (int st = 128; st > 0; st >>= 1) {
    if (threadIdx.x < st) {
      smin[threadIdx.x] = min(smin[threadIdx.x], smin[threadIdx.x + st]);
      smax[threadIdx.x] = max(smax[threadIdx.x], smax[threadIdx.x + st]);
    }
    __syncthreads();
  }
  if (threadIdx.x == 0) {
    unsigned* mm = (unsigned*)(ws + OFF_MNMX);
    atomicMin(&mm[b*2],   smin[0]);
    atomicMax(&mm[b*2+1], smax[0]);
  }
}

// 1x1 conv via WMMA f32 16x16x4: out[o,pix] = sum_c w[o,c]*feat[c,pix] (+bias)
// one wave per 16-pixel tile; also accumulates GroupNorm sum/sumsq stats
__global__ void k_conv(const float* __restrict__ feat,
                       const float* __restrict__ cw,
                       const float* __restrict__ cb,
                       float* ws) {
  int lane = threadIdx.x;           // 0..31
  int hi = lane >> 4, lo = lane & 15;
  int bb   = blockIdx.x >> 10;      // / (HW/16)
  int pix0 = (blockIdx.x & 1023) << 4;

  const float* fB = feat + (size_t)bb * C_ * HW_ + pix0 + lo;
  v8f acc = {};
  #pragma unroll
  for (int kk = 0; kk < 16; ++kk) {
    int c0 = kk * 4 + hi * 2;
    v2f a, bf;
    a.x  = cw[lo * C_ + c0];
    a.y  = cw[lo * C_ + c0 + 1];
    bf.x = fB[(size_t)c0 * HW_];
    bf.y = fB[(size_t)(c0 + 1) * HW_];
    acc = __builtin_amdgcn_wmma_f32_16x16x4_f32(
        false, a, false, bf, (short)0, acc, false, false);
  }

  float sg[2]  = {0.f, 0.f};
  float ssg[2] = {0.f, 0.f};
  float* rr = ws + OFF_RRAW;
  #pragma unroll
  for (int r = 0; r < 8; ++r) {
    int o = r + 8 * hi;                    // output channel
    float v = acc[r] + cb[o];
    rr[((size_t)bb * 16 + o) * HW_ + pix0 + lo] = v;
    sg[r >> 2]  += v;
    ssg[r >> 2] += v * v;
  }
  // butterfly reduce across the 16 lanes of each half-wave
  #pragma unroll
  for (int m = 1; m < 16; m <<= 1) {
    sg[0]  += __shfl_xor(sg[0],  m, 32);
    sg[1]  += __shfl_xor(sg[1],  m, 32);
    ssg[0] += __shfl_xor(ssg[0], m, 32);
    ssg[1] += __shfl_xor(ssg[1], m, 32);
  }
  if (lo == 0) {
    int g0 = 2 * hi;                       // hi=0 -> groups 0,1 ; hi=1 -> 2,3
    atomicAdd(&ws[OFF_STATS + bb*8 + g0*2 + 0],     sg[0]);
    atomicAdd(&ws[OFF_STATS + bb*8 + g0*2 + 1],     ssg[0]);
    atomicAdd(&ws[OFF_STATS + bb*8 + (g0+1)*2 + 0], sg[1]);
    atomicAdd(&ws[OFF_STATS + bb*8 + (g0+1)*2 + 1], ssg[1]);
  }
}

__global__ void k_fin(float* ws) {
  int t = threadIdx.x;
  unsigned* mm = (unsigned*)(ws + OFF_MNMX);
  if (t < B_) {
    float mn = decf(mm[t*2]), mx = decf(mm[t*2+1]);
    ws[OFF_MNF + t*2]     = mn;
    ws[OFF_MNF + t*2 + 1] = 1.f / (mx - mn + EPS_);
  }
  if (t < 32) {
    int b = t >> 2, g = t & 3;
    float s = ws[OFF_STATS + b*8 + g*2];
    float q = ws[OFF_STATS + b*8 + g*2 + 1];
    const float inv = 1.f / (4.f * HW_);
    float mu  = s * inv;
    float var = q * inv - mu * mu;
    ws[OFF_GNF + b*8 + g*2]     = mu;
    ws[OFF_GNF + b*8 + g*2 + 1] = rsqrtf(var + EPS_);
  }
}

// pooled[b, ch, ph, pw] = mean_{8x8}( sf_norm + relu(GN(conv)) )
__global__ void k_pool(const float* __restrict__ gamma,
                       const float* __restrict__ beta, float* ws) {
  int p = blockIdx.x * 256 + threadIdx.x;   // B*16*16*16 = 32768
  int b  = p >> 12;
  int rem = p & 4095;
  int ch = rem >> 8;
  int ph = (rem >> 4) & 15, pw = rem & 15;
  int g = ch >> 2;
  float mu = ws[OFF_GNF + b*8 + g*2], rs = ws[OFF_GNF + b*8 + g*2 + 1];
  float mn = ws[OFF_MNF + b*2],       sc = ws[OFF_MNF + b*2 + 1];
  float ga = gamma[ch], be = beta[ch];
  const float* rr = ws + OFF_RRAW + ((size_t)b * 16 + ch) * HW_;
  const float* sf = ws + OFF_SF + (size_t)b * HW_;
  float acc = 0.f;
  for (int ih = 0; ih < 8; ++ih) {
    int hh = ph * 8 + ih;
    #pragma unroll
    for (int iw = 0; iw < 8; ++iw) {
      int hw = hh * W_ + pw * 8 + iw;
      float rn = fmaxf((rr[hw] - mu) * rs * ga + be, 0.f);
      acc += (sf[hw] - mn) * sc + rn;
    }
  }
  ws[OFF_POOL + p] = acc * (1.f / 64.f);
}

// MLP head + 11x11 normalized tap table, one block per batch
__global__ void k_mlp(const float* __restrict__ w1,
                      const float* __restrict__ w2, float* ws) {
  __shared__ float flat[4096];
  __shared__ float red[128];
  __shared__ float sdil;
  int b = blockIdx.x, t = threadIdx.x;
  for (int i = t; i < 4096; i += 128) flat[i] = ws[OFF_POOL + b*4096 + i];
  __syncthreads();

  const float* w1r = w1 + (size_t)t * 4096;
  float acc = 0.f;
  for (int k = 0; k < 4096; k += 4)
    acc += flat[k]*w1r[k] + flat[k+1]*w1r[k+1]
         + flat[k+2]*w1r[k+2] + flat[k+3]*w1r[k+3];
  red[t] = fmaxf(acc, 0.f) * w2[t];
  __syncthreads();
  for (int st = 64; st > 0; st >>= 1) {
    if (t < st) red[t] += red[t + st];
    __syncthreads();
  }
  if (t == 0) {
    float sig = 1.f / (1.f + expf(-red[0]));
    sdil = (float)MD_ * sig;
    ws[OFF_DIL + b] = sdil;
  }
  __syncthreads();

  float e = 0.f;
  if (t < K_) {
    float dx = (float)(t / 11 - MD_), dy = (float)(t % 11 - MD_);
    e = expf(-sqrtf(dx*dx + dy*dy) / sdil);
  }
  red[t] = e;
  __syncthreads();
  for (int st = 64; st > 0; st >>= 1) {
    if (t < st) red[t] += red[t + st];
    __syncthreads();
  }
  if (t < K_) ws[OFF_WKER + b*K_ + t] = e / red[0];
}

__global__ void k_copy(const float4* __restrict__ src, float4* __restrict__ dst) {
  size_t i = (size_t)blockIdx.x * 256 + threadIdx.x;
  dst[i] = src[i];
}

// fused nearest-cell scatter (weight 1 folded into center tap) + weighted splat
__global__ void k_splat(const float* __restrict__ x,
                        const float* __restrict__ xyz,
                        const float* __restrict__ ws, float* out) {
  __shared__ float xv[C_];
  int bn = blockIdx.x;              // B*N blocks
  int b = bn >> 12, n = bn & 4095;
  int t = threadIdx.x;
  if (t < C_) xv[t] = x[((size_t)b * N_ + n) * C_ + t];
  float u = xyz[((size_t)b * N_ + n) * 3 + 0];
  float v = xyz[((size_t)b * N_ + n) * 3 + 1];
  int xc = (int)(u * (float)(W_ - 1));
  int yc = (int)(v * (float)(H_ - 1));
  __syncthreads();
  if (t >= K_) return;

  float wt = ws[OFF_WKER + b*K_ + t];
  if (t == 60) wt += 1.f;           // center tap carries the nearest-cell scatter
  int dx = t / 11 - MD_, dy = t % 11 - MD_;
  int xn = min(max(xc + dx, 0), W_ - 1);
  int yn = min(max(yc + dy, 0), H_ - 1);
  int lin = yn * W_ + xn;
  float* ob = out + (size_t)b * C_ * HW_ + lin;
  #pragma unroll 4
  for (int c = 0; c < C_; ++c)
    atomicAdd(ob + (size_t)c * HW_, xv[c] * wt);
}

extern "C" void kernel_launch(void* const* d_in, const int* in_sizes, int n_in,
                              void* d_out, int out_size, void* d_ws, size_t ws_size,
                              hipStream_t stream) {
  const float* x        = (const float*)d_in[0];
  const float* xyz      = (const float*)d_in[1];
  const float* feature  = (const float*)d_in[2];
  const float* conv_w   = (const float*)d_in[3];
  const float* conv_b   = (const float*)d_in[4];
  const float* gn_gamma = (const float*)d_in[5];
  const float* gn_beta  = (const float*)d_in[6];
  const float* mlp_w1   = (const float*)d_in[7];
  const float* mlp_w2   = (const float*)d_in[8];
  float* out = (float*)d_out;
  float* ws  = (float*)d_ws;

  k_init <<<1,     64,  0, stream>>>(ws);
  k_sf   <<<512,   256, 0, stream>>>(feature, ws);
  k_conv <<<8192,  32,  0, stream>>>(feature, conv_w, conv_b, ws);
  k_fin  <<<1,     32,  0, stream>>>(ws);
  k_pool <<<128,   256, 0, stream>>>(gn_gamma, gn_beta, ws);
  k_mlp  <<<B_,    128, 0, stream>>>(mlp_w1, mlp_w2, ws);
  k_copy <<<8192,  256, 0, stream>>>((const float4*)feature, (float4*)out);
  k_splat<<<B_*N_, 128, 0, stream>>>(x, xyz, ws, out);
}